// QBlockGeLU_73847667687987
// MI455X (gfx1250) — compile-verified
//
#include <hip/hip_runtime.h>
#include <math.h>
#include <stdint.h>

// ---------------------------------------------------------------------------
// QBlockGeLU for MI455X (gfx1250):
//   - async global->LDS staging (ASYNCcnt / CDNA5 async-copy path)
//   - wave32-per-32x32-block, lane-per-row; absmax via wave shuffles
//   - rintf (v_rndne) quantization, reciprocal-multiply dequant
//   - non-temporal b128 stores (output stream > 192MB L2)
// Memory-bound: 512 MB @ 23.3 TB/s ~= 22 us floor.
// ---------------------------------------------------------------------------

typedef float v4f __attribute__((ext_vector_type(4)));
typedef int   v4i __attribute__((ext_vector_type(4)));

#define MAT_N      8192
#define TILE_ROWS  32
#define TILE_COLS  256          // 8 blocks of 32 per workgroup
#define LDS_STRIDE 260          // 256 + 4 floats pad: 1040B % 256B = 16B -> minimal-conflict b128 reads
#define QMAX_F     127.0f
#define EPS_F      1e-8f

// ---- CDNA5 async global->LDS b128 copy (tracked by ASYNCcnt) --------------
__device__ __forceinline__ void async_load_b128(const float* gptr,
                                                uint32_t lds_byte_off,
                                                const float* lds_base) {
  // low 32 bits of a flat shared pointer == group-segment byte offset
  uint32_t laddr = (uint32_t)(uintptr_t)lds_base + lds_byte_off;
#if defined(__has_builtin) && __has_builtin(__builtin_amdgcn_global_load_async_to_lds_b128)
  __builtin_amdgcn_global_load_async_to_lds_b128(
      (__attribute__((address_space(1))) v4i*)(uintptr_t)gptr,
      (__attribute__((address_space(3))) v4i*)(uint32_t)laddr,
      /*imm offset*/ 0, /*cpol*/ 0);
#else
  uint64_t ga = (uint64_t)(uintptr_t)gptr;
  asm volatile("global_load_async_to_lds_b128 %0, %1, off"
               :
               : "v"(laddr), "v"(ga)
               : "memory");
#endif
}

__device__ __forceinline__ void wait_async0() {
#if defined(__has_builtin) && __has_builtin(__builtin_amdgcn_s_wait_asynccnt)
  __builtin_amdgcn_s_wait_asynccnt(0);
#else
  asm volatile("s_wait_asynccnt 0" ::: "memory");
#endif
}

// ---- wave32 butterfly max --------------------------------------------------
__device__ __forceinline__ float wave_max(float v) {
#pragma unroll
  for (int m = 16; m >= 1; m >>= 1)
    v = fmaxf(v, __shfl_xor(v, m, 32));
  return v;
}

__device__ __forceinline__ float gelu_exact(float x) {
  return 0.5f * x * (1.0f + erff(x * 0.70710678118654752440f));
}

// round-half-even (v_rndne_f32) matches jnp.round; clamp to +-127
__device__ __forceinline__ float qdq1(float x, float s, float rs) {
  float q = rintf(x * rs);
  q = fminf(fmaxf(q, -QMAX_F), QMAX_F);
  return q * s;
}

extern "C" __global__ __launch_bounds__(256)
void QBlockGeLU_73847667687987_kernel(const float* __restrict__ x,
                                      float* __restrict__ out) {
  __shared__ float tile[TILE_ROWS * LDS_STRIDE];   // 33,280 B of 320 KB WGP LDS

  const int tilesX = MAT_N / TILE_COLS;            // 32 tiles per row band
  const int wg     = blockIdx.x;                   // 8192 workgroups total
  const int trow   = wg / tilesX;
  const int tcol   = wg - trow * tilesX;
  const int t      = threadIdx.x;                  // 0..255

  const size_t gbase =
      (size_t)trow * TILE_ROWS * MAT_N + (size_t)tcol * TILE_COLS;

  // ---- Stage 1: async-copy the 32x256 fp32 tile into LDS (coalesced) -----
  // 2048 float4-chunks; chunk c: row = c/64, col4 = c%64. One wave issue
  // covers 512 contiguous bytes of a row.
#pragma unroll
  for (int i = 0; i < 8; ++i) {
    const int c   = t + 256 * i;
    const int row = c >> 6;
    const int c4  = c & 63;
    const float* g = x + gbase + (size_t)row * MAT_N + (c4 << 2);
    const uint32_t loff =
        (uint32_t)((row * LDS_STRIDE + (c4 << 2)) * sizeof(float));
    async_load_b128(g, loff, tile);
  }
  wait_async0();
  __syncthreads();

  // ---- Stage 2: one wave per 32x32 block, one lane per block row ----------
  const int wv   = t >> 5;   // block index within tile (0..7)
  const int lane = t & 31;   // row within block

  v4f v[8];
  float m = 0.0f;
#pragma unroll
  for (int k = 0; k < 8; ++k) {
    v[k] = *(const v4f*)&tile[lane * LDS_STRIDE + wv * 32 + k * 4];
    m = fmaxf(m, fmaxf(fmaxf(fabsf(v[k].x), fabsf(v[k].y)),
                       fmaxf(fabsf(v[k].z), fabsf(v[k].w))));
  }
  m = wave_max(m);
  const float s1 = fmaxf(m * (1.0f / QMAX_F), EPS_F);
  const float r1 = 1.0f / s1;

  // quant-dequant -> exact GELU -> track activation absmax
  float m2 = 0.0f;
#pragma unroll
  for (int k = 0; k < 8; ++k) {
    v4f y;
    y.x = gelu_exact(qdq1(v[k].x, s1, r1));
    y.y = gelu_exact(qdq1(v[k].y, s1, r1));
    y.z = gelu_exact(qdq1(v[k].z, s1, r1));
    y.w = gelu_exact(qdq1(v[k].w, s1, r1));
    v[k] = y;
    m2 = fmaxf(m2, fmaxf(fmaxf(fabsf(y.x), fabsf(y.y)),
                         fmaxf(fabsf(y.z), fabsf(y.w))));
  }
  m2 = wave_max(m2);
  const float s2 = fmaxf(m2 * (1.0f / QMAX_F), EPS_F);
  const float r2 = 1.0f / s2;

  // ---- Stage 3: requantize + streaming (non-temporal) b128 stores ---------
  float* orow = out + gbase + (size_t)lane * MAT_N + wv * 32;
#pragma unroll
  for (int k = 0; k < 8; ++k) {
    v4f y;
    y.x = qdq1(v[k].x, s2, r2);
    y.y = qdq1(v[k].y, s2, r2);
    y.z = qdq1(v[k].z, s2, r2);
    y.w = qdq1(v[k].w, s2, r2);
    __builtin_nontemporal_store(y, (v4f*)(orow + 4 * k));
  }
}

extern "C" void kernel_launch(void* const* d_in, const int* in_sizes, int n_in,
                              void* d_out, int out_size, void* d_ws, size_t ws_size,
                              hipStream_t stream) {
  (void)in_sizes; (void)n_in; (void)d_ws; (void)ws_size; (void)out_size;
  const float* x = (const float*)d_in[0];   // 8192x8192 fp32
  // d_in[1] is B == 32, baked into the kernel's tiling.
  float* out = (float*)d_out;

  const int tiles = (MAT_N / TILE_ROWS) * (MAT_N / TILE_COLS);  // 256*32 = 8192
  QBlockGeLU_73847667687987_kernel<<<dim3(tiles), dim3(256), 0, stream>>>(x, out);
}